// CVML_27599459845026
// MI455X (gfx1250) — compile-verified
//
#include <hip/hip_runtime.h>
#include <hip/hip_bf16.h>

typedef __attribute__((ext_vector_type(2))) float v2f;
typedef __attribute__((ext_vector_type(8))) float v8f;

#define MS_OFF 33562624L   // 128*64*4097

// ---------------- mask (channel-max) kernels ----------------
__global__ void cvml_mask_grd(const float* __restrict__ g, float* __restrict__ out) {
    int idx = blockIdx.x * blockDim.x + threadIdx.x;      // 128*256
    int b = idx >> 8, hw = idx & 255;
    const float* p = g + (long)b * 131072 + hw;           // 512*256
    float m = -3.402823466e+38f;
    for (int c = 0; c < 512; ++c) m = fmaxf(m, p[(long)c * 256]);
    out[idx] = m;
}

__global__ void cvml_mask_pts(const float* __restrict__ g, float* __restrict__ out) {
    int idx = blockIdx.x * blockDim.x + threadIdx.x;      // 128*1024
    int b = idx >> 10, pp = idx & 1023;
    const float* p = g + (long)b * 524288 + pp;           // 512*1024
    float m = -3.402823466e+38f;
    for (int c = 0; c < 512; ++c) m = fmaxf(m, p[(long)c * 1024]);
    out[idx] = m;
}

__global__ void cvml_mask_sat(const float* __restrict__ g, float* __restrict__ out) {
    int idx = blockIdx.x * blockDim.x + threadIdx.x;      // 8192*4
    int bp = idx >> 2, i = idx & 3;
    int b = bp >> 6, sy = (bp >> 3) & 7, sx = bp & 7;
    int iy = i >> 1, ix = i & 1;
    const float* p = g + (long)b * 131072 + (long)(sy * 2 + iy) * 16 + (sx * 2 + ix);
    float m = -3.402823466e+38f;
    for (int c = 0; c < 512; ++c) m = fmaxf(m, p[(long)c * 256]);
    out[idx] = m;
}

// ---------------- tiny sat MLP (4 -> 2 -> 4, x8 heads) ----------------
__global__ void cvml_sat_mlp(const float* __restrict__ msk,
                             const float* __restrict__ w1, const float* __restrict__ b1,
                             const float* __restrict__ w2, const float* __restrict__ b2,
                             float* __restrict__ wt) {
    int bp = blockIdx.x * blockDim.x + threadIdx.x;       // 8192
    float mk[4];
#pragma unroll
    for (int i = 0; i < 4; ++i) mk[i] = msk[bp * 4 + i];
    float m[2][8];
#pragma unroll
    for (int j = 0; j < 2; ++j)
#pragma unroll
        for (int d = 0; d < 8; ++d) {
            float s = b1[j * 8 + d];
#pragma unroll
            for (int i = 0; i < 4; ++i) s += mk[i] * w1[i * 16 + j * 8 + d];
            m[j][d] = s;
        }
#pragma unroll
    for (int i = 0; i < 4; ++i)
#pragma unroll
        for (int d = 0; d < 8; ++d) {
            float s = b2[i * 8 + d];
#pragma unroll
            for (int j = 0; j < 2; ++j) s += m[j][d] * w2[j * 32 + i * 8 + d];
            wt[bp * 32 + i * 8 + d] = s;
        }
}

// ---------------- generic strided batched WMMA f32 GEMM (compile-time strides) -----
// C[bz][m][n] = sum_k A[bz][m][k] * B[bz][k][n] (+ bias[bz][n])
// Wave computes MT 16x16 tiles stacked in M, sharing the B fragment per K step.
// K stepped by 4 (V_WMMA_F32_16X16X4_F32). A frag: lane r -> row; lanes>=16 hold K+2.
template<long A_RS, long A_CS, long A_BS,
         long B_RS, long B_CS, long B_BS,
         long C_RS, long C_BS,
         long BIAS_NS, long BIAS_BS, bool HAS_BIAS,
         int K, int NLOG, int MT>
__launch_bounds__(128)
__global__ void cvml_gemm_wmma(const float* __restrict__ A, const float* __restrict__ B,
                               float* __restrict__ C, const float* __restrict__ bias) {
    int lane = threadIdx.x;          // 0..31
    int g = lane >> 4, r = lane & 15;
    int mt0 = (blockIdx.x * 4 + threadIdx.y) * MT;        // first m-tile of this wave
    int n0 = blockIdx.y * 16;
    long bz = blockIdx.z;
    const float* Ab = A + bz * A_BS;
    const float* Bb = B + bz * B_BS;
    int nl = n0 + r; if (nl > NLOG - 1) nl = NLOG - 1;    // clamp: loads in-bounds, EXEC all-1
    const float* arow[MT];
#pragma unroll
    for (int t = 0; t < MT; ++t)
        arow[t] = Ab + (long)((mt0 + t) * 16 + r) * A_RS;
    v8f zero = {};
    v8f acc[MT];
#pragma unroll
    for (int t = 0; t < MT; ++t) acc[t] = zero;
    for (int k = 0; k < K; k += 4) {
        int ka = k + 2 * g;
        v2f bv;
        bv.x = Bb[(long)ka * B_RS + (long)nl * B_CS];
        bv.y = Bb[(long)(ka + 1) * B_RS + (long)nl * B_CS];
#pragma unroll
        for (int t = 0; t < MT; ++t) {
            v2f av;
            av.x = arow[t][(long)ka * A_CS];
            av.y = arow[t][(long)(ka + 1) * A_CS];
            acc[t] = __builtin_amdgcn_wmma_f32_16x16x4_f32(false, av, false, bv,
                                                           (short)0, acc[t], false, false);
        }
    }
    int nn = n0 + r;
    if (nn < NLOG) {
        float bval = HAS_BIAS ? bias[bz * BIAS_BS + (long)nn * BIAS_NS] : 0.0f;
        float* ccol = C + bz * C_BS + nn;
#pragma unroll
        for (int t = 0; t < MT; ++t)
#pragma unroll
            for (int v = 0; v < 8; ++v)
                ccol[(long)((mt0 + t) * 16 + v + 8 * g) * C_RS] = acc[t][v] + bval;
    }
}

// ---------------- sat patches: (512x4)@(4x8) per patch, single-K WMMA ----------------
__launch_bounds__(128)
__global__ void cvml_sat_patch_wmma(const float* __restrict__ sat,
                                    const float* __restrict__ wt,
                                    float* __restrict__ outAll) {
    int lane = threadIdx.x, g = lane >> 4, r = lane & 15;
    int m0 = (blockIdx.x * 4 + threadIdx.y) * 16;         // channel tile (0..496)
    int bp = blockIdx.y;                                   // 0..8191
    int b = bp >> 6, sy = (bp >> 3) & 7, sx = bp & 7;
    long base = (long)b * 131072 + (long)(sy * 2) * 16 + (sx * 2);
    int k0 = 2 * g, k1 = k0 + 1;
    int c = m0 + r;
    v2f av, bv;
    av.x = sat[base + (long)c * 256 + (k0 >> 1) * 16 + (k0 & 1)];
    av.y = sat[base + (long)c * 256 + (k1 >> 1) * 16 + (k1 & 1)];
    int nd = r & 7;                                        // clamp n to 8 heads
    bv.x = wt[bp * 32 + k0 * 8 + nd];
    bv.y = wt[bp * 32 + k1 * 8 + nd];
    v8f acc = {};
    acc = __builtin_amdgcn_wmma_f32_16x16x4_f32(false, av, false, bv,
                                                (short)0, acc, false, false);
    if (r < 8) {
        float* o = outAll + (long)bp * 4097 + 1 + r;       // raw sat_global into cost_map
#pragma unroll
        for (int v = 0; v < 8; ++v)
            o[(long)(m0 + v + 8 * g) * 8] = acc[v];
    }
}

// ---------------- L2 norm of 4096-length rows ----------------
__global__ void cvml_l2norm_rows(const float* __restrict__ in, float* __restrict__ out) {
    __shared__ float red[256];
    long row = blockIdx.x;
    const float* p = in + row * 4096;
    float ss = 0.f;
    for (int t = threadIdx.x; t < 4096; t += 256) { float v = p[t]; ss += v * v; }
    red[threadIdx.x] = ss; __syncthreads();
    for (int s = 128; s > 0; s >>= 1) {
        if (threadIdx.x < s) red[threadIdx.x] += red[threadIdx.x + s];
        __syncthreads();
    }
    float scale = 1.0f / fmaxf(sqrtf(red[0]), 1e-12f);
    for (int t = threadIdx.x; t < 4096; t += 256) out[row * 4096 + t] = p[t] * scale;
}

// ---------------- finalize: normalize sat_global in place, compute ms ----------------
__global__ void cvml_sat_finalize(float* __restrict__ outAll,
                                  const float* __restrict__ gn,
                                  const float* __restrict__ pn) {
    __shared__ float red[256];
    int bp = blockIdx.x, b = bp >> 6;
    float* cm = outAll + (long)bp * 4097;
    float* sv = cm + 1;
    float ss = 0.f;
    for (int t = threadIdx.x; t < 4096; t += 256) { float v = sv[t]; ss += v * v; }
    red[threadIdx.x] = ss; __syncthreads();
    for (int s = 128; s > 0; s >>= 1) {
        if (threadIdx.x < s) red[threadIdx.x] += red[threadIdx.x + s];
        __syncthreads();
    }
    float scale = 1.0f / fmaxf(sqrtf(red[0]), 1e-12f);
    __syncthreads();
    const float* g = gn + (long)b * 4096;
    const float* p = pn + (long)b * 4096;
    float dot = 0.f;
    for (int t = threadIdx.x; t < 4096; t += 256) {
        float v = sv[t] * scale; sv[t] = v;
        dot += (g[t] + p[t]) * v;
    }
    red[threadIdx.x] = dot; __syncthreads();
    for (int s = 128; s > 0; s >>= 1) {
        if (threadIdx.x < s) red[threadIdx.x] += red[threadIdx.x + s];
        __syncthreads();
    }
    if (threadIdx.x == 0) {
        float ms = red[0];
        cm[0] = ms;
        outAll[MS_OFF + bp] = ms;
    }
}

extern "C" void kernel_launch(void* const* d_in, const int* in_sizes, int n_in,
                              void* d_out, int out_size, void* d_ws, size_t ws_size,
                              hipStream_t stream) {
    const float* grd_local = (const float*)d_in[0];   // (128,512,8,32)
    const float* points    = (const float*)d_in[1];   // (128,512,1024,1)
    const float* sat_local = (const float*)d_in[2];   // (128,512,16,16)
    const float* grd_w1 = (const float*)d_in[3];      // (256,128,8)
    const float* grd_b1 = (const float*)d_in[4];      // (1,128,8)
    const float* grd_w2 = (const float*)d_in[5];      // (128,256,8)
    const float* grd_b2 = (const float*)d_in[6];      // (1,256,8)
    const float* sat_w1 = (const float*)d_in[7];      // (4,2,8)
    const float* sat_b1 = (const float*)d_in[8];
    const float* sat_w2 = (const float*)d_in[9];      // (2,4,8)
    const float* sat_b2 = (const float*)d_in[10];
    const float* pc_w1  = (const float*)d_in[11];     // (1024,512,8)
    const float* pc_b1  = (const float*)d_in[12];
    const float* pc_w2  = (const float*)d_in[13];     // (512,1024,8)
    const float* pc_b2  = (const float*)d_in[14];
    float* out = (float*)d_out;
    float* ws  = (float*)d_ws;

    float* mask_grd = ws;                // 32768
    float* mask_pts = ws + 32768;        // 131072
    float* mask_sat = ws + 163840;       // 32768
    float* m_grd    = ws + 196608;       // [8][128][128]
    float* grd_wt   = ws + 327680;       // [8][128][256]
    float* m_pc     = ws + 589824;       // [8][128][512]
    float* pc_wt    = ws + 1114112;      // [8][128][1024]
    float* sat_wt   = ws + 2162688;      // [8192][4][8]
    float* grd_raw  = ws + 2424832;      // [128][512][8]
    float* pts_raw  = ws + 2949120;      // [128][512][8]
    float* g_norm   = ws + 3473408;      // [128][4096]
    float* p_norm   = ws + 3997696;      // [128][4096]

    dim3 blk(32, 4);

    cvml_mask_grd<<<128, 256, 0, stream>>>(grd_local, mask_grd);
    cvml_mask_pts<<<512, 256, 0, stream>>>(points, mask_pts);
    cvml_mask_sat<<<128, 256, 0, stream>>>(sat_local, mask_sat);
    cvml_sat_mlp<<<32, 256, 0, stream>>>(mask_sat, sat_w1, sat_b1, sat_w2, sat_b2, sat_wt);

    // grd MLP: m = mask(128,256) @ w1_d(256,128) + b1   (z = 8 heads)
    cvml_gemm_wmma<256, 1, 0,   1024, 8, 1,   128, 16384,   8, 1, true,  256, 128, 2>
        <<<dim3(1, 8, 8), blk, 0, stream>>>(mask_grd, grd_w1, m_grd, grd_b1);
    // grd_wt = m_d(128,128) @ w2_d(128,256) + b2
    cvml_gemm_wmma<128, 1, 16384,   2048, 8, 1,   256, 32768,   8, 1, true,  128, 256, 2>
        <<<dim3(1, 16, 8), blk, 0, stream>>>(m_grd, grd_w2, grd_wt, grd_b2);
    // grd_global raw: per-batch (512,256)@(256,8); z = 128, N padded to 16
    cvml_gemm_wmma<256, 1, 131072,   1, 32768, 256,   8, 4096,   0, 0, false,  256, 8, 2>
        <<<dim3(4, 1, 128), blk, 0, stream>>>(grd_local, grd_wt, grd_raw, nullptr);

    // pc MLP: m = mask(128,1024) @ w1_d(1024,512) + b1
    cvml_gemm_wmma<1024, 1, 0,   4096, 8, 1,   512, 65536,   8, 1, true,  1024, 512, 2>
        <<<dim3(1, 32, 8), blk, 0, stream>>>(mask_pts, pc_w1, m_pc, pc_b1);
    // pc_wt = m_d(128,512) @ w2_d(512,1024) + b2
    cvml_gemm_wmma<512, 1, 65536,   8192, 8, 1,   1024, 131072,   8, 1, true,  512, 1024, 2>
        <<<dim3(1, 64, 8), blk, 0, stream>>>(m_pc, pc_w2, pc_wt, pc_b2);
    // pts raw: per-batch (512,1024)@(1024,8)
    cvml_gemm_wmma<1024, 1, 524288,   1, 131072, 1024,   8, 4096,   0, 0, false,  1024, 8, 2>
        <<<dim3(4, 1, 128), blk, 0, stream>>>(points, pc_wt, pts_raw, nullptr);

    // sat patches -> raw sat_global straight into d_out cost_map[...,1:]
    cvml_sat_patch_wmma<<<dim3(8, 8192), blk, 0, stream>>>(sat_local, sat_wt, out);

    cvml_l2norm_rows<<<128, 256, 0, stream>>>(grd_raw, g_norm);
    cvml_l2norm_rows<<<128, 256, 0, stream>>>(pts_raw, p_norm);
    cvml_sat_finalize<<<8192, 256, 0, stream>>>(out, g_norm, p_norm);
}